// BoundaryLoss_74861279969285
// MI455X (gfx1250) — compile-verified
//
#include <hip/hip_runtime.h>
#include <math.h>

// BoundaryLoss for MI455X (gfx1250), wave32.
// B=8, H=W=256. f32 throughout. WMMA f32_16x16x4_f32 used for the per-batch
// weighted-sum reductions (f32 inputs + f32 accumulate -> bit-compatible precision).

typedef __attribute__((ext_vector_type(2))) float v2f;
typedef __attribute__((ext_vector_type(8))) float v8f;

#define NB 8
#define NH 256
#define NW 256
#define NHW (NH * NW)
#define BIGD 1.0e4f

// ---------------------------------------------------------------- boundary
// boundary = tb && !erode3x3(tb)  (erosion with 3x3 ones, border padded False)
__global__ void k_boundary(const int* __restrict__ tgt, float* __restrict__ buf) {
    int idx = blockIdx.x * 256 + threadIdx.x;        // < NB*NHW
    int b = idx / NHW, r = idx % NHW;
    int i = r / NW, j = r % NW;
    const int* t = tgt + b * NHW;
    bool tb = (t[r] != 0);
    bool er = false;
    if (i > 0 && i < NH - 1 && j > 0 && j < NW - 1) {
        er = true;
#pragma unroll
        for (int di = -1; di <= 1; ++di)
#pragma unroll
            for (int dj = -1; dj <= 1; ++dj)
                er = er && (t[(i + di) * NW + (j + dj)] != 0);
    }
    buf[idx] = (tb && !er) ? 1.0f : 0.0f;
}

// ---------------------------------------------------------------- vertical EDT
// In-place per-column scan: buf holds boundary (1/0) in, f (vertical dist, clamped
// to BIGD) out. One thread per (batch, column); accesses coalesced across columns.
__global__ void k_vscan(float* __restrict__ buf) {
    int b = blockIdx.x;             // NB blocks
    int j = threadIdx.x;            // NW threads
    float* col = buf + b * NHW + j;
    float nd = BIGD;
    for (int i = 0; i < NH; ++i) {
        float v = col[i * NW];
        nd = (v != 0.0f) ? 0.0f : fminf(nd + 1.0f, BIGD);
        col[i * NW] = nd;           // 0 marks boundary for the upward pass
    }
    float nu = BIGD;
    for (int i = NH - 1; i >= 0; --i) {
        float v = col[i * NW];
        nu = (v == 0.0f) ? 0.0f : fminf(nu + 1.0f, BIGD);
        col[i * NW] = fminf(v, nu);
    }
}

// ---------------------------------------------------------------- horizontal min-plus
// d[i,j] = sqrt( min_j' f[i,j']^2 + (j-j')^2 ), in-place per row.
// One block per (b,i): f^2 row in LDS, uniform-broadcast inner loop of fma+min.
__global__ void k_hpass(float* __restrict__ buf) {
    __shared__ float f2[NW];
    int row = blockIdx.x;           // NB*NH blocks
    int j = threadIdx.x;            // NW threads
    float* rp = buf + row * NW;
    float f = rp[j];
    f2[j] = f * f;
    __syncthreads();
    float best = 3.4e38f;
#pragma unroll 4
    for (int jp = 0; jp < NW; ++jp) {
        float dj = (float)(j - jp);
        best = fminf(best, __fmaf_rn(dj, dj, f2[jp]));
    }
    rp[j] = sqrtf(best);
}

// ---------------------------------------------------------------- per-batch max(d)
__global__ void k_max(const float* __restrict__ dbuf, float* __restrict__ maxd) {
    __shared__ float sm[256];
    int b = blockIdx.x, t = threadIdx.x;
    const float* p = dbuf + b * NHW;
    float m = 0.0f;
    for (int k = t; k < NHW; k += 256) m = fmaxf(m, p[k]);
    sm[t] = m;
    __syncthreads();
    for (int s = 128; s > 0; s >>= 1) {
        if (t < s) sm[t] = fmaxf(sm[t], sm[t + s]);
        __syncthreads();
    }
    if (t == 0) maxd[b] = sm[0];
}

// ---------------------------------------------------------------- WMMA weighted sums
// Per batch: s_d = sum(d), s_pd = sum(sigmoid(logit)*d), accumulated with
// V_WMMA_F32_16X16X4_F32: D += A(16x4) * ones(4x16). Every column of D carries
// the grand total, so wave-sum(D) == 16 * S (exact /16). f32 in, f32 accumulate.
__global__ void k_sums(const float* __restrict__ logits, const float* __restrict__ dbuf,
                       float* __restrict__ sd, float* __restrict__ spd) {
    __shared__ float wden[8], wnum[8];
    int b = blockIdx.x;                       // NB blocks of 256 threads = 8 waves
    int wave = threadIdx.x >> 5;
    int lane = threadIdx.x & 31;
    const float* dp = dbuf + b * NHW + wave * 8192;
    const float* lp = logits + b * NHW + wave * 8192;

    v2f ones; ones.x = 1.0f; ones.y = 1.0f;
    v8f cd = {};                              // accumulates sum(d)
    v8f cn = {};                              // accumulates sum(pred*d)

    for (int t = 0; t < 128; ++t) {           // 128 * 64 = 8192 elems per wave
        int o = t * 64 + lane;
        float d0 = dp[o], d1 = dp[o + 32];
        float x0 = lp[o], x1 = lp[o + 32];
        float p0 = 1.0f / (1.0f + expf(-x0));
        float p1 = 1.0f / (1.0f + expf(-x1));
        v2f ad; ad.x = d0;      ad.y = d1;
        v2f an; an.x = p0 * d0; an.y = p1 * d1;
        // D(16x16,f32) = A(16x4,f32) x ones(4x16,f32) + C
        cd = __builtin_amdgcn_wmma_f32_16x16x4_f32(false, ad, false, ones,
                                                   (short)0, cd, false, false);
        cn = __builtin_amdgcn_wmma_f32_16x16x4_f32(false, an, false, ones,
                                                   (short)0, cn, false, false);
    }

    float ld = 0.0f, ln = 0.0f;
#pragma unroll
    for (int k = 0; k < 8; ++k) { ld += cd[k]; ln += cn[k]; }
#pragma unroll
    for (int m = 16; m > 0; m >>= 1) {
        ld += __shfl_xor(ld, m, 32);
        ln += __shfl_xor(ln, m, 32);
    }
    if (lane == 0) {
        wden[wave] = ld * (1.0f / 16.0f);     // 16 identical columns -> exact /16
        wnum[wave] = ln * (1.0f / 16.0f);
    }
    __syncthreads();
    if (threadIdx.x == 0) {
        float a = 0.0f, c = 0.0f;
        for (int w = 0; w < 8; ++w) { a += wden[w]; c += wnum[w]; }
        sd[b] = a;
        spd[b] = c;
    }
}

// ---------------------------------------------------------------- final combine
// per[b] = (spd*inv) / (sd*inv + 1e-7), inv = 1/(max+1e-7); out = mean(per)
__global__ void k_final(const float* __restrict__ maxd, const float* __restrict__ sd,
                        const float* __restrict__ spd, float* __restrict__ out) {
    float acc = 0.0f;
    for (int b = 0; b < NB; ++b) {
        float inv = 1.0f / (maxd[b] + 1e-7f);
        float num = spd[b] * inv;
        float den = sd[b] * inv + 1e-7f;
        acc += num / den;
    }
    out[0] = acc * (1.0f / (float)NB);
}

extern "C" void kernel_launch(void* const* d_in, const int* in_sizes, int n_in,
                              void* d_out, int out_size, void* d_ws, size_t ws_size,
                              hipStream_t stream) {
    const float* logits = (const float*)d_in[0];   // [8,1,256,256] f32
    const int*   target = (const int*)d_in[1];     // [8,1,256,256] i32
    float* out = (float*)d_out;                    // scalar f32

    float* buf  = (float*)d_ws;                    // NB*NHW floats (2 MB), reused in place
    float* maxd = buf + NB * NHW;                  // 8
    float* sd   = maxd + NB;                       // 8
    float* spd  = sd + NB;                         // 8

    k_boundary<<<NB * NHW / 256, 256, 0, stream>>>(target, buf);
    k_vscan   <<<NB, NW, 0, stream>>>(buf);
    k_hpass   <<<NB * NH, NW, 0, stream>>>(buf);
    k_max     <<<NB, 256, 0, stream>>>(buf, maxd);
    k_sums    <<<NB, 256, 0, stream>>>(logits, buf, sd, spd);
    k_final   <<<1, 1, 0, stream>>>(maxd, sd, spd, out);
}